// SparseMHA_42142219108794
// MI455X (gfx1250) — compile-verified
//
#include <hip/hip_runtime.h>
#include <hip/hip_bf16.h>

#define HIDDEN   256
#define HEADS    8
#define HEAD_DIM 32
#define NNODES   50000   // 3125 * 16 : row tiles divide evenly, no OOB guards needed
#define NEDGES   800000

typedef __attribute__((ext_vector_type(2))) float v2f;
typedef __attribute__((ext_vector_type(8))) float v8f;
typedef __attribute__((ext_vector_type(4))) int   v4i;

// Optional gfx1250 async global->LDS path (direct LDS fill, tracked by ASYNCcnt)
#if defined(__has_builtin)
#  if __has_builtin(__builtin_amdgcn_global_load_async_to_lds_b128)
#    define HAVE_ASYNC_LDS 1
#  endif
#  if __has_builtin(__builtin_amdgcn_s_wait_asynccnt)
#    define HAVE_WAIT_ASYNC 1
#  endif
#endif

// ---- monotone float<->uint encoding for atomic max over signed floats ----
__device__ __forceinline__ unsigned enc_f32(float f) {
    unsigned u = __float_as_uint(f);
    return (u & 0x80000000u) ? ~u : (u | 0x80000000u);
}
__device__ __forceinline__ float dec_f32(unsigned u) {
    unsigned b = (u & 0x80000000u) ? (u & 0x7FFFFFFFu) : ~u;
    return __uint_as_float(b);
}
#define ENC_NEG_INF 0x007FFFFFu   // enc_f32(-inf)

// ============================================================================
// Y = X @ W^T + bias (optionally * scale), tiled 16x16 per wave using
// V_WMMA_F32_16X16X4_F32 (full fp32 — matches the fp32 reference).
// Block: 256 threads = 8 waves. grid.x tiles rows by 16 (nrows % 16 == 0),
// grid.y (=2) picks which 8 of the 16 column tiles this block's waves own.
// ============================================================================
#define LDSW 260   // padded row stride (floats); 260 % 64 == 4 -> conflict-free
__global__ __launch_bounds__(256)
void gat_wmma_gemm_xT(const float* __restrict__ X, const float* __restrict__ W,
                      const float* __restrict__ bias, float* __restrict__ Y,
                      float scale) {
    __shared__ float tile[16 * LDSW];
    const int rowbase = blockIdx.x * 16;
    const int tid = threadIdx.x;

    // Cooperative load of the 16x256 X tile. 1024 float4 slots, 4 per thread.
    // 16B alignment holds in LDS too: LDSW*4 = 1040 bytes = 65*16.
    #pragma unroll
    for (int j = 0; j < 4; ++j) {
        int f4 = tid + 256 * j;          // 0..1023
        int r  = f4 >> 6;                // 64 float4 per row
        int c4 = f4 & 63;
        const float* src = X + (size_t)(rowbase + r) * HIDDEN + c4 * 4;
        float*       dst = &tile[r * LDSW + c4 * 4];
#ifdef HAVE_ASYNC_LDS
        __builtin_amdgcn_global_load_async_to_lds_b128(
            (__attribute__((address_space(1))) v4i*)src,
            (__attribute__((address_space(3))) v4i*)dst, 0, 0);
#else
        float4 val = *(const float4*)src;
        dst[0] = val.x; dst[1] = val.y; dst[2] = val.z; dst[3] = val.w;
#endif
    }
#ifdef HAVE_ASYNC_LDS
#  ifdef HAVE_WAIT_ASYNC
    __builtin_amdgcn_s_wait_asynccnt(0);
#  else
    asm volatile("s_wait_asynccnt 0x0" ::: "memory");
#  endif
#endif
    __syncthreads();

    const int lane  = tid & 31;
    const int wave  = tid >> 5;
    const int ctile = blockIdx.y * 8 + wave;        // 0..15
    const int col   = ctile * 16 + (lane & 15);     // output column of this lane
    const int khalf = (lane >> 4) * 2;              // K sub-pair: 0 or 2

    const float* __restrict__ wrow = W + (size_t)col * HIDDEN;  // B[k][col] = W[col][k]
    __builtin_prefetch(wrow, 0, 0);                 // global_prefetch_b8

    const float* __restrict__ arow = &tile[(lane & 15) * LDSW]; // A row m = lane&15

    v8f acc = {};
    #pragma unroll 8
    for (int k0 = 0; k0 < HIDDEN; k0 += 4) {
        v2f a, b;
        a.x = arow[k0 + khalf];         // A: VGPR0 = K=khalf, VGPR1 = K=khalf+1
        a.y = arow[k0 + khalf + 1];
        b.x = wrow[k0 + khalf];         // B mirrored layout
        b.y = wrow[k0 + khalf + 1];
        acc = __builtin_amdgcn_wmma_f32_16x16x4_f32(
                  false, a, false, b, (short)0, acc, false, false);
    }

    // C/D layout: VGPR i -> M = i (lanes 0-15) or M = 8+i (lanes 16-31), N = lane&15
    const float bvs   = bias[col] * scale;          // (acc+b)*s == acc*s + b*s
    const int   mbase = (lane >> 4) * 8;
    float* __restrict__ yp = Y + (size_t)(rowbase + mbase) * HIDDEN + col;
    #pragma unroll
    for (int i = 0; i < 8; ++i) {
        yp[(size_t)i * HIDDEN] = fmaf(acc[i], scale, bvs);
    }
}

// ============================================================================
// Init per-(node,head) segment-max (encoded -inf) and segment-sum (0)
// ============================================================================
__global__ __launch_bounds__(256)
void gat_init_mz(unsigned* __restrict__ menc, float* __restrict__ z) {
    int i = blockIdx.x * blockDim.x + threadIdx.x;
    if (i < NNODES * HEADS) { menc[i] = ENC_NEG_INF; z[i] = 0.f; }
}

__global__ __launch_bounds__(256)
void gat_zero(float* __restrict__ p, long n) {
    long i = (long)blockIdx.x * blockDim.x + threadIdx.x;
    if (i < n) p[i] = 0.f;
}

// ============================================================================
// SDDMM: s[e,h] = sum_d q[row[e], d*8+h] * k[col[e], d*8+h]; segment max by row.
// One thread per (edge, head). Gathers hit L2 (q,k,v = 153.6MB < 192MB L2).
// ============================================================================
__global__ __launch_bounds__(256)
void gat_sddmm(const float* __restrict__ q, const float* __restrict__ k,
               const int* __restrict__ row, const int* __restrict__ col,
               float* __restrict__ sbuf, unsigned* __restrict__ menc) {
    long gtid = (long)blockIdx.x * blockDim.x + threadIdx.x;
    if (gtid >= (long)NEDGES * HEADS) return;
    int e = (int)(gtid >> 3);
    int h = (int)(gtid & 7);
    int r = row[e], c = col[e];
    const float* __restrict__ qp = q + (size_t)r * HIDDEN + h;
    const float* __restrict__ kp = k + (size_t)c * HIDDEN + h;
    float s = 0.f;
    #pragma unroll
    for (int d = 0; d < HEAD_DIM; ++d) s += qp[d * HEADS] * kp[d * HEADS];
    sbuf[(size_t)e * HEADS + h] = s;
    atomicMax(&menc[(size_t)r * HEADS + h], enc_f32(s));
}

// ============================================================================
// e = exp(s - m[row]); segment sum z by row.
// ============================================================================
__global__ __launch_bounds__(256)
void gat_expz(const int* __restrict__ row, float* __restrict__ sbuf,
              const unsigned* __restrict__ menc, float* __restrict__ z) {
    long gtid = (long)blockIdx.x * blockDim.x + threadIdx.x;
    if (gtid >= (long)NEDGES * HEADS) return;
    int e = (int)(gtid >> 3);
    int h = (int)(gtid & 7);
    int r = row[e];
    size_t idx = (size_t)e * HEADS + h;
    float m  = dec_f32(menc[(size_t)r * HEADS + h]);
    float ex = __expf(sbuf[idx] - m);
    sbuf[idx] = ex;
    atomicAdd(&z[(size_t)r * HEADS + h], ex);
}

// ============================================================================
// SPMM: out[row, d*8+h] += (e/z) * v[col, d*8+h].  One wave per edge; lane owns
// channels {lane + 32j}, all of which share head h = lane & 7. Atomic f32 adds
// land in L2 (51.2MB accumulator is L2-resident).
// ============================================================================
__global__ __launch_bounds__(256)
void gat_spmm(const int* __restrict__ row, const int* __restrict__ col,
              const float* __restrict__ sbuf, const float* __restrict__ z,
              const float* __restrict__ v, float* __restrict__ accum) {
    int lane = threadIdx.x & 31;
    int wave = threadIdx.x >> 5;
    int e = blockIdx.x * 8 + wave;
    if (e >= NEDGES) return;
    int r = row[e], c = col[e];
    int h = lane & 7;
    float attn = sbuf[(size_t)e * HEADS + h] / z[(size_t)r * HEADS + h];
    const float* __restrict__ vp = v + (size_t)c * HIDDEN;
    float* __restrict__ op = accum + (size_t)r * HIDDEN;
    #pragma unroll
    for (int j = 0; j < 8; ++j) {
        int cc = lane + 32 * j;              // cc % 8 == lane & 7 == h
        atomicAdd(&op[cc], attn * vp[cc]);
    }
}

// ============================================================================
// Host launcher
// ============================================================================
extern "C" void kernel_launch(void* const* d_in, const int* in_sizes, int n_in,
                              void* d_out, int out_size, void* d_ws, size_t ws_size,
                              hipStream_t stream) {
    const float* h   = (const float*)d_in[0];
    const int*   row = (const int*)d_in[1];
    const int*   col = (const int*)d_in[2];
    const float* Wq  = (const float*)d_in[3];
    const float* bq  = (const float*)d_in[4];
    const float* Wk  = (const float*)d_in[5];
    const float* bk  = (const float*)d_in[6];
    const float* Wv  = (const float*)d_in[7];
    const float* bv  = (const float*)d_in[8];
    const float* Wo  = (const float*)d_in[9];
    const float* bo  = (const float*)d_in[10];
    float* out = (float*)d_out;

    // Workspace layout (floats). Q buffer is reused as the SPMM accumulator
    // (Q is dead after SDDMM).  Total ~182 MB.
    const size_t NH = (size_t)NNODES * HIDDEN;
    float*    q    = (float*)d_ws;
    float*    k    = q + NH;
    float*    v    = k + NH;
    float*    sbuf = v + NH;                                   // E*H scores
    float*    z    = sbuf + (size_t)NEDGES * HEADS;            // N*H sums
    unsigned* menc = (unsigned*)(z + (size_t)NNODES * HEADS);  // N*H encoded max

    const float scaling = 0.17677669529663689f;  // HEAD_DIM^-0.5

    dim3 blk(256);
    dim3 gemmGrid(NNODES / 16, 2);               // NNODES % 16 == 0
    const long eh = (long)NEDGES * HEADS;

    // 1) init segment max/sum tables
    gat_init_mz<<<(NNODES * HEADS + 255) / 256, blk, 0, stream>>>(menc, z);

    // 2) fp32-WMMA projections: q = (h Wq^T + bq)*scale; k, v unscaled
    gat_wmma_gemm_xT<<<gemmGrid, blk, 0, stream>>>(h, Wq, bq, q, scaling);
    gat_wmma_gemm_xT<<<gemmGrid, blk, 0, stream>>>(h, Wk, bk, k, 1.0f);
    gat_wmma_gemm_xT<<<gemmGrid, blk, 0, stream>>>(h, Wv, bv, v, 1.0f);

    // 3) SDDMM + segment max
    gat_sddmm<<<(unsigned)((eh + 255) / 256), blk, 0, stream>>>(q, k, row, col, sbuf, menc);

    // 4) exp(s - m) + segment sum
    gat_expz<<<(unsigned)((eh + 255) / 256), blk, 0, stream>>>(row, sbuf, menc, z);

    // 5) SPMM into accumulator (reuse q's space, zero it first)
    float* accum = q;
    gat_zero<<<(unsigned)((NH + 255) / 256), blk, 0, stream>>>(accum, (long)NH);
    gat_spmm<<<(NEDGES + 7) / 8, blk, 0, stream>>>(row, col, sbuf, z, v, accum);

    // 6) fp32-WMMA output projection straight into d_out
    gat_wmma_gemm_xT<<<gemmGrid, blk, 0, stream>>>(accum, Wo, bo, out, 1.0f);
}